// GaussianKernelBiasingDensity_37684043055751
// MI455X (gfx1250) — compile-verified
//
#include <hip/hip_runtime.h>

#define B_N 16384
#define M_N 16384
#define INV_VAR 100.0f
#define EPS_W 0.01f
#define LOG2E 1.4426950408889634f
#define S_CHUNKS 8
#define COLS_PER_CHUNK (M_N / S_CHUNKS)          /* 2048 */
#define TILES_PER_CHUNK (COLS_PER_CHUNK / 16)    /* 128 */

typedef __attribute__((ext_vector_type(2))) float v2f;
typedef __attribute__((ext_vector_type(8))) float v8f;

// Per-mixture constant, pre-scaled to base-2 exponent domain:
// c2[m] = log2e * ( -0.5*invvar*(mz^2+mt^2) + log(w[m]+eps) )
__global__ void gkb_prep(const float* __restrict__ means,
                         const float* __restrict__ weights,
                         float* __restrict__ cvals) {
  int m = blockIdx.x * blockDim.x + threadIdx.x;
  if (m < M_N) {
    float mz = means[2 * m + 0];
    float mt = means[2 * m + 1];
    cvals[m] = LOG2E * (-0.5f * INV_VAR * (mz * mz + mt * mt)
                        + logf(weights[m] + EPS_W));
  }
}

// Each wave: one 16-row batch tile x one 2048-column mixture chunk.
// pot2 = log2e*(cross + a_row + c_col); D = A*B + C folds a_row in via the
// WMMA accumulator; base-2 scaling is baked into A, arow, cvals so the inner
// loop is: wmma -> add cl -> v_exp_f32 -> accumulate.
__global__ __launch_bounds__(256) void gkb_main(const float* __restrict__ z,
                                                const float* __restrict__ t,
                                                const float* __restrict__ means,
                                                const float* __restrict__ cvals,
                                                float* __restrict__ partial) {
  const int lane    = threadIdx.x & 31;
  const int wave    = (blockIdx.x * blockDim.x + threadIdx.x) >> 5;
  const int rowTile = wave / S_CHUNKS;
  const int chunk   = wave % S_CHUNKS;
  const int rowBase = rowTile * 16;
  const int l15     = lane & 15;
  const int hi      = lane >> 4;   // 0 = lanes 0-15, 1 = lanes 16-31

  // Loop-invariant row constant, in the WMMA C layout:
  // C VGPR i <-> row (i + hi*8), column (lane & 15).
  v8f cacc;
#pragma unroll
  for (int i = 0; i < 8; ++i) {
    int r = rowBase + i + hi * 8;
    float zz = z[r], tt = t[r];
    cacc[i] = LOG2E * (-0.5f * INV_VAR * (zz * zz + tt * tt));
  }

  // A (16x4 f32): lanes 0-15 hold (K0,K1) = log2e*invvar*(z,t) for row
  // rowBase+l15; lanes 16-31 are the K2,K3 zero padding.
  const float s = INV_VAR * LOG2E;
  v2f afrag;
  afrag[0] = hi ? 0.0f : z[rowBase + l15] * s;
  afrag[1] = hi ? 0.0f : t[rowBase + l15] * s;

  float acc[8];
#pragma unroll
  for (int i = 0; i < 8; ++i) acc[i] = 0.0f;

  const int colStart = chunk * COLS_PER_CHUNK;
  const float2* mcols = (const float2*)means;  // means rows are (mz, mt) pairs

  for (int j = 0; j < TILES_PER_CHUNK; ++j) {
    const int col = colStart + j * 16 + l15;
    const float2 mc = mcols[col];
    const float cl  = cvals[col];   // column constant, needed by every lane

    // B (4x16 f32): lanes 0-15 = rows (K0,K1) = (mz,mt); lanes 16-31 = 0 pad.
    v2f bfrag;
    bfrag[0] = hi ? 0.0f : mc.x;
    bfrag[1] = hi ? 0.0f : mc.y;

    // D = cross2 + arow2 (accumulator folds the row constant in)
    v8f d = __builtin_amdgcn_wmma_f32_16x16x4_f32(
        false, afrag, false, bfrag, (short)0, cacc, false, false);

#pragma unroll
    for (int i = 0; i < 8; ++i) {
      acc[i] += __builtin_amdgcn_exp2f(d[i] + cl);
    }
  }

  // acc[i] on lane L holds the partial sum for row (i + hi*8) restricted to
  // columns == (L&15) mod 16. Reduce across the 16 lanes of each half-wave.
#pragma unroll
  for (int i = 0; i < 8; ++i) {
    float v = acc[i];
    v += __shfl_xor(v, 1);
    v += __shfl_xor(v, 2);
    v += __shfl_xor(v, 4);
    v += __shfl_xor(v, 8);
    if (l15 == 0) partial[chunk * B_N + rowBase + i + hi * 8] = v;
  }
}

// Deterministic fixed-order combine of the S_CHUNKS partial sums.
__global__ void gkb_reduce(const float* __restrict__ partial,
                           float* __restrict__ out) {
  int r = blockIdx.x * blockDim.x + threadIdx.x;
  if (r < B_N) {
    float sum = 0.0f;
#pragma unroll
    for (int c = 0; c < S_CHUNKS; ++c) sum += partial[c * B_N + r];
    out[r] = -sum;
  }
}

extern "C" void kernel_launch(void* const* d_in, const int* in_sizes, int n_in,
                              void* d_out, int out_size, void* d_ws, size_t ws_size,
                              hipStream_t stream) {
  const float* z       = (const float*)d_in[0];  // (B,1)
  const float* t       = (const float*)d_in[1];  // (B,1)
  const float* means   = (const float*)d_in[2];  // (128,128,2)
  const float* weights = (const float*)d_in[3];  // (128,128,1)
  float* out = (float*)d_out;

  float* cvals   = (float*)d_ws;        // M_N floats
  float* partial = cvals + M_N;         // S_CHUNKS * B_N floats

  gkb_prep<<<M_N / 256, 256, 0, stream>>>(means, weights, cvals);

  // (B_N/16) row tiles * S_CHUNKS chunks = 8192 waves; 8 waves (256 thr)/block
  gkb_main<<<(B_N / 16) * S_CHUNKS / 8, 256, 0, stream>>>(z, t, means, cvals, partial);

  gkb_reduce<<<B_N / 256, 256, 0, stream>>>(partial, out);
}